// GraphLanzcosConvolution_13846974562558
// MI455X (gfx1250) — compile-verified
//
#include <hip/hip_runtime.h>
#include <hip/hip_bf16.h>
#include <math.h>

#define N_NODES 50000
#define FDIM    128
#define DDIM    128
#define KSUP    4
#define NEDGE   800000
#define EPSN    1e-12f

typedef __attribute__((ext_vector_type(2))) float v2f;
typedef __attribute__((ext_vector_type(8))) float v8f;

__device__ __forceinline__ void atomAddF(float* p, float v) {
    // lowers to global_atomic_add_f32 (no return) on gfx1250
    unsafeAtomicAdd(p, v);
}

// ---------------- elementwise helpers ----------------

__global__ void k_zero(float* __restrict__ p, int n) {
    int i = blockIdx.x * blockDim.x + threadIdx.x;
    if (i < n) p[i] = 0.0f;
}

// out[f] += sum over 128-row chunk of x[r,f]^2   (blockDim = 128)
__global__ void k_colsumsq(const float* __restrict__ x, float* __restrict__ out, int n) {
    int f  = threadIdx.x;
    int r0 = blockIdx.x * 128;
    int r1 = r0 + 128; if (r1 > n) r1 = n;
    float s = 0.0f;
    for (int r = r0; r < r1; ++r) {
        float v = x[(long)r * FDIM + f];
        s += v * v;
    }
    atomAddF(&out[f], s);
}

// v0 = x * rsqrt(max(colnorm2, eps))
__global__ void k_scale_rsqrt(const float* __restrict__ x, const float* __restrict__ nrm2,
                              float* __restrict__ v0, int total) {
    int i = blockIdx.x * blockDim.x + threadIdx.x;
    if (i >= total) return;
    int f = i & (FDIM - 1);
    v0[i] = x[i] * rsqrtf(fmaxf(nrm2[f], EPSN));
}

// ---------------- SpMV: w[row,:] += val * v[col,:] ----------------
// one 32-lane group per edge, 4 contiguous floats per lane (512B per edge)
__global__ void k_spmv(const float* __restrict__ v, const int* __restrict__ erow,
                       const int* __restrict__ ecol, const float* __restrict__ eval,
                       float* __restrict__ w) {
    long tid = (long)blockIdx.x * blockDim.x + threadIdx.x;
    int e = (int)(tid >> 5);
    if (e >= NEDGE) return;
    int j = ((int)tid & 31) * 4;
    float a = eval[e];
    const float4 vv = *(const float4*)&v[(long)ecol[e] * FDIM + j];
    float* wp = &w[(long)erow[e] * FDIM + j];
    atomAddF(wp + 0, a * vv.x);
    atomAddF(wp + 1, a * vv.y);
    atomAddF(wp + 2, a * vv.z);
    atomAddF(wp + 3, a * vv.w);
}

// out[f] += sum over 128-row chunk of a[r,f]*b[r,f]   (blockDim = 128)
__global__ void k_coldot(const float* __restrict__ a, const float* __restrict__ b,
                         float* __restrict__ out, int n) {
    int f  = threadIdx.x;
    int r0 = blockIdx.x * 128;
    int r1 = r0 + 128; if (r1 > n) r1 = n;
    float s = 0.0f;
    for (int r = r0; r < r1; ++r) {
        long idx = (long)r * FDIM + f;
        s += a[idx] * b[idx];
    }
    atomAddF(&out[f], s);
}

// new_v = w - alpha*v_prev - beta_prev*v_prev2  (in place over w), accumulate beta2
__global__ void k_update(float* __restrict__ w, const float* __restrict__ vp,
                         const float* __restrict__ vpp, const float* __restrict__ alpha,
                         const float* __restrict__ beta2_prev, int has_prev,
                         float* __restrict__ beta2_out, int n) {
    int f  = threadIdx.x;
    float al = alpha[f];
    float bp = has_prev ? sqrtf(beta2_prev[f]) : 0.0f;
    int r0 = blockIdx.x * 128;
    int r1 = r0 + 128; if (r1 > n) r1 = n;
    float s = 0.0f;
    for (int r = r0; r < r1; ++r) {
        long idx = (long)r * FDIM + f;
        float nv = w[idx] - al * vp[idx] - bp * vpp[idx];
        w[idx] = nv;
        s += nv * nv;
    }
    atomAddF(&beta2_out[f], s);
}

// v /= sqrt(beta2[f])
__global__ void k_scale_div(float* __restrict__ v, const float* __restrict__ beta2, int total) {
    int i = blockIdx.x * blockDim.x + threadIdx.x;
    if (i >= total) return;
    int f = i & (FDIM - 1);
    v[i] = v[i] / sqrtf(beta2[f]);
}

// ---------------- Final GEMM with fp32 WMMA ----------------
// out[n,d] = relu( sum_{k,f} vbuf[k][n][f] * W[k][f][d] )
// = [N x 512] @ [512 x 128] with K-loop of V_WMMA_F32_16X16X4_F32.
// Block: 256 threads = 8 waves; wave w owns columns [16w, 16w+16); block owns 16 rows.
__global__ __launch_bounds__(256)
void k_gemm_wmma(const float* __restrict__ vbuf, const float* __restrict__ wts,
                 float* __restrict__ out) {
    const int wave    = threadIdx.x >> 5;
    const int lane    = threadIdx.x & 31;
    const int half    = lane >> 4;      // 0: lanes 0-15 (K+0,K+1), 1: lanes 16-31 (K+2,K+3)
    const int l16     = lane & 15;
    const int rowBase = blockIdx.x * 16;
    const int colBase = wave * 16;

    const int rowA = rowBase + l16;     // A fragment: M = lane%16
    const int dcol = colBase + l16;     // B fragment: N = lane%16

    v8f c = {};
    for (int kk = 0; kk < KSUP * FDIM; kk += 4) {
        const int kf = kk + half * 2;   // this lane's first K index (4 K per wmma)
        const int ko = kf >> 7;         // which support matrix
        const int f  = kf & 127;        // feature index inside it

        // A: 16x4 fp32, 2 VGPRs per lane: (K = kf, kf+1)
        v2f a;
        {
            const float2 t = *(const float2*)&vbuf[((long)ko * N_NODES + rowA) * FDIM + f];
            a.x = t.x; a.y = t.y;
        }
        // B: 4x16 fp32, 2 VGPRs per lane: rows K = kf, kf+1 at column dcol
        v2f b;
        {
            const float* wp = &wts[((long)ko * FDIM + f) * DDIM + dcol];
            b.x = wp[0];
            b.y = wp[DDIM];
        }
        c = __builtin_amdgcn_wmma_f32_16x16x4_f32(
                /*neg_a=*/false, a, /*neg_b=*/false, b,
                /*c_mod=*/(short)0, c, /*reuse_a=*/false, /*reuse_b=*/false);
    }

    // C/D layout: VGPR r -> M = r (lanes 0-15) or r+8 (lanes 16-31), N = lane%16
#pragma unroll
    for (int r = 0; r < 8; ++r) {
        int m = rowBase + r + half * 8;
        out[(long)m * DDIM + dcol] = fmaxf(c[r], 0.0f);
    }
}

// ---------------- launch ----------------

extern "C" void kernel_launch(void* const* d_in, const int* in_sizes, int n_in,
                              void* d_out, int out_size, void* d_ws, size_t ws_size,
                              hipStream_t stream) {
    const float* x         = (const float*)d_in[0];
    const float* edge_vals = (const float*)d_in[1];
    const float* weights   = (const float*)d_in[2];
    const int*   edge_row  = (const int*)d_in[3];
    const int*   edge_col  = (const int*)d_in[4];
    float*       out       = (float*)d_out;

    // workspace layout (floats): [0,128) norm2 | [128,512) alpha[3] | [512,896) beta2[3]
    // | padding to 1024 | vbuf[4][N][F]
    float* wsf   = (float*)d_ws;
    float* norm2 = wsf;
    float* alpha = wsf + 128;
    float* beta2 = wsf + 512;
    float* vbuf  = wsf + 1024;

    const int  total   = N_NODES * FDIM;            // 6,400,000
    const int  eblocks = (total + 255) / 256;
    const int  cblocks = (N_NODES + 127) / 128;     // 391

    // zero scalar reduction area
    k_zero<<<4, 256, 0, stream>>>(wsf, 1024);

    // v0 = l2_normalize(x) per column
    k_colsumsq<<<cblocks, 128, 0, stream>>>(x, norm2, N_NODES);
    k_scale_rsqrt<<<eblocks, 256, 0, stream>>>(x, norm2, vbuf, total);

    // 3 Lanczos steps
    for (int i = 0; i < 3; ++i) {
        float* vcur  = vbuf + (long)i * total;
        float* vnext = vbuf + (long)(i + 1) * total;     // doubles as w
        const float* vpp = (i >= 1) ? (vbuf + (long)(i - 1) * total) : vbuf;

        k_zero<<<eblocks, 256, 0, stream>>>(vnext, total);
        {
            long spmvThreads = (long)NEDGE * 32;
            int  blocks = (int)((spmvThreads + 255) / 256);   // 100000
            k_spmv<<<blocks, 256, 0, stream>>>(vcur, edge_row, edge_col, edge_vals, vnext);
        }
        k_coldot<<<cblocks, 128, 0, stream>>>(vnext, vcur, alpha + i * 128, N_NODES);
        k_update<<<cblocks, 128, 0, stream>>>(vnext, vcur, vpp, alpha + i * 128,
                                              (i >= 1) ? (beta2 + (i - 1) * 128) : beta2,
                                              (i >= 1) ? 1 : 0,
                                              beta2 + i * 128, N_NODES);
        k_scale_div<<<eblocks, 256, 0, stream>>>(vnext, beta2 + i * 128, total);
    }

    // out = relu( sum_k Vs[k] @ W[k] )  via fp32 WMMA
    k_gemm_wmma<<<N_NODES / 16, 256, 0, stream>>>(vbuf, weights, out);
}